// SectionMeanPool_74483322847799
// MI455X (gfx1250) — compile-verified
//
#include <hip/hip_runtime.h>
#include <hip/hip_bf16.h>

typedef __attribute__((ext_vector_type(2))) float v2f;
typedef __attribute__((ext_vector_type(8))) float v8f;

#define BB 32
#define SS 7
#define TT 2048
#define HH 768
#define EPSF 1e-9f

// ---------------------------------------------------------------------------
// Kernel 1: per-(b,s) mask counts -> reciprocal (clipped at EPS), into d_ws
// ---------------------------------------------------------------------------
__global__ void SectionMeanPool_count_kernel(const int* __restrict__ masks,
                                             float* __restrict__ recip) {
    const int s = blockIdx.x;   // 0..6
    const int b = blockIdx.y;   // 0..31
    const int* m = masks + ((size_t)b * SS + s) * TT;

    int sum = 0;
    for (int t = threadIdx.x; t < TT; t += blockDim.x) sum += m[t];

    __shared__ int red[256];
    red[threadIdx.x] = sum;
    __syncthreads();
    for (int off = 128; off > 0; off >>= 1) {
        if ((int)threadIdx.x < off) red[threadIdx.x] += red[threadIdx.x + off];
        __syncthreads();
    }
    if (threadIdx.x == 0) {
        float c = (float)red[0];
        recip[b * SS + s] = 1.0f / fmaxf(c, EPSF);
    }
}

// ---------------------------------------------------------------------------
// Kernel 2: masked-sum GEMM via V_WMMA_F32_16X16X4_F32 (exact fp32 math).
// Grid: x = H/16 tiles (48), y = batch (32). Block: 256 threads = 8 waves.
// Each wave covers a T-chunk of 256 (64 WMMAs, K=4); partial 16x16 tiles are
// reduced across waves in LDS; wave 0 scales & stores rows 0..6.
//
// Branch-free inner loop: invalid A rows (7..15) read a clamped-but-valid
// address and are zeroed with a 0.0/1.0 multiply; the prefetch address is
// clamped with min() instead of guarded with a branch.
//
// A (16x4 fp32, ISA layout): lanes 0-15 -> M=lane, K=0(V0),1(V1);
//                            lanes 16-31 -> M=lane-16, K=2(V0),3(V1).
// B (4x16 fp32, analogous half-split): lanes 0-15 -> N=lane, K=0(V0),1(V1);
//                            lanes 16-31 -> K=2(V0),3(V1).
// C/D (16x16 fp32, ISA layout): VGPR r -> rows r (lanes 0-15) / r+8
//                            (lanes 16-31), N = lane%16.
// ---------------------------------------------------------------------------
__global__ void SectionMeanPool_wmma_kernel(const float* __restrict__ x,
                                            const int* __restrict__ masks,
                                            const float* __restrict__ recip,
                                            float* __restrict__ out) {
    const int hb   = blockIdx.x * 16;      // H tile base
    const int b    = blockIdx.y;           // batch
    const int wave = threadIdx.x >> 5;     // 0..7
    const int lane = threadIdx.x & 31;
    const int half = lane >> 4;            // 0 | 1
    const int l16  = lane & 15;

    const float* __restrict__ xb = x + (size_t)b * TT * HH;
    const int*   __restrict__ mb = masks + (size_t)b * SS * TT;

    v8f acc = {};
    const int   t0     = wave * (TT / 8);          // 256-wide T chunk
    const int   koff   = half * 2;                 // K offset for this half
    const int   arow   = l16;                      // logical A row
    const int   arow_c = (arow < SS) ? arow : (SS - 1);   // clamped (valid)
    const float avalid = (arow < SS) ? 1.0f : 0.0f;       // zero padding
    const int* __restrict__ mrow = mb + arow_c * TT;
    const float* __restrict__ xcol = xb + hb + l16;

#pragma unroll 4
    for (int i = 0; i < 64; ++i) {
        const int tk = t0 + i * 4 + koff;

        // A tile: unconditional 8B mask load, zeroed by multiply for pad rows
        const int2 mv = *(const int2*)&mrow[tk];
        v2f a;
        a.x = avalid * (float)mv.x;
        a.y = avalid * (float)mv.y;

        // B tile: hidden rows tk, tk+1 at column hb+l16
        v2f bm;
        bm.x = xcol[(size_t)tk       * HH];
        bm.y = xcol[(size_t)(tk + 1) * HH];

        // Branch-free streaming prefetch, 4 WMMA-steps (16 rows) ahead
        const int pfr = (tk + 16 < TT) ? (tk + 16) : (TT - 1);
        __builtin_prefetch(&xcol[(size_t)pfr * HH], 0, 3);

        // D = A*B + C, fp32 in / fp32 accumulate (matches reference exactly)
        acc = __builtin_amdgcn_wmma_f32_16x16x4_f32(
            /*neg_a=*/false, a, /*neg_b=*/false, bm,
            /*c_mod=*/(short)0, acc, /*reuse_a=*/false, /*reuse_b=*/false);
    }

    // Cross-wave reduction of the 8 partial 16x16 tiles in LDS
    __shared__ float red[8 * 256];
    float* myred = red + wave * 256;
#pragma unroll
    for (int r = 0; r < 8; ++r) myred[r * 32 + lane] = acc[r];
    __syncthreads();

    if (wave == 0) {
#pragma unroll
        for (int r = 0; r < 8; ++r) {
            float v = 0.0f;
#pragma unroll
            for (int w = 0; w < 8; ++w) v += red[w * 256 + r * 32 + lane];
            const int row = r + half * 8;  // C/D layout: rows r / r+8
            if (row < SS) {
                out[((size_t)b * SS + row) * HH + hb + l16] =
                    v * recip[b * SS + row];
            }
        }
    }
}

// ---------------------------------------------------------------------------
// Launch
// ---------------------------------------------------------------------------
extern "C" void kernel_launch(void* const* d_in, const int* in_sizes, int n_in,
                              void* d_out, int out_size, void* d_ws, size_t ws_size,
                              hipStream_t stream) {
    const float* hidden = (const float*)d_in[0];   // (32, 2048, 768) fp32
    const int*   masks  = (const int*)d_in[1];     // (32, 7, 2048) int32
    float* out   = (float*)d_out;                  // (32, 7*768) fp32
    float* recip = (float*)d_ws;                   // 32*7 floats scratch

    // Pass 1: reciprocal counts
    dim3 cgrid(SS, BB);
    SectionMeanPool_count_kernel<<<cgrid, 256, 0, stream>>>(masks, recip);

    // Pass 2: WMMA masked-sum + scale
    dim3 ggrid(HH / 16, BB);
    SectionMeanPool_wmma_kernel<<<ggrid, 256, 0, stream>>>(hidden, masks, recip, out);
}